// EdgeGenerator_4569845202979
// MI455X (gfx1250) — compile-verified
//
#include <hip/hip_runtime.h>
#include <hip/hip_bf16.h>

typedef __attribute__((ext_vector_type(2))) float v2f;
typedef __attribute__((ext_vector_type(8))) float v8f;

#define D_DIM 128
// sigmoid(log(s)+eps) == s / (s + exp(-eps)), eps = 1e-5
#define EXP_NEG_EPS 0.99999000005f

// ---------------------------------------------------------------------------
// Kernel 0: zero the 128-float btw accumulator living in workspace.
// ---------------------------------------------------------------------------
__global__ void eg_zero_btw(float* __restrict__ btw) {
    int t = threadIdx.x;
    if (t < D_DIM) btw[t] = 0.0f;
}

// ---------------------------------------------------------------------------
// Kernel 1: btw[d] = sum_e embed[edge_idx[e,1], d] * weight[e]
// One edge per wave per iteration: 32 lanes x float4 = full 512B row,
// coalesced gather from the L2-resident table. Per-lane accumulators own
// fixed dims; block reduction via LDS ds_add_f32, then 128 global atomics.
// ---------------------------------------------------------------------------
__global__ void eg_btw_accum(const float* __restrict__ embed,
                             const float* __restrict__ weight,
                             const int*   __restrict__ eidx,
                             float*       __restrict__ btw,
                             int nEdges) {
    __shared__ float lred[D_DIM];
    const int tid  = threadIdx.x;
    const int lane = tid & 31;
    const int wave = tid >> 5;

    if (tid < D_DIM) lred[tid] = 0.0f;
    __syncthreads();

    const int gwave  = blockIdx.x * (blockDim.x >> 5) + wave;
    const int nwaves = gridDim.x * (blockDim.x >> 5);

    float4 acc = make_float4(0.f, 0.f, 0.f, 0.f);
    for (int e = gwave; e < nEdges; e += nwaves) {
        const int   idx = eidx[2 * e + 1];          // column 1: B rows
        const float w   = weight[e];
        const float4 v  = *(const float4*)(embed + (size_t)idx * D_DIM + lane * 4);
        acc.x = fmaf(w, v.x, acc.x);
        acc.y = fmaf(w, v.y, acc.y);
        acc.z = fmaf(w, v.z, acc.z);
        acc.w = fmaf(w, v.w, acc.w);
    }

    // LDS float atomics (ds_add_f32): every wave's lane L owns dims 4L..4L+3.
    atomicAdd(&lred[lane * 4 + 0], acc.x);
    atomicAdd(&lred[lane * 4 + 1], acc.y);
    atomicAdd(&lred[lane * 4 + 2], acc.z);
    atomicAdd(&lred[lane * 4 + 3], acc.w);
    __syncthreads();

    if (tid < D_DIM) atomicAdd(&btw[tid], lred[tid]);   // global_atomic_add_f32
}

// ---------------------------------------------------------------------------
// Kernel 2: score[e] = dot(embed[edge_idx[e,0]], btw); out = s/(s+exp(-eps)).
// Per wave: tiles of 16 edges. K=128 reduced in the matrix unit with
// V_WMMA_F32_16X16X4_F32 (32 K-steps). A-layout (32-bit 16x4): lanes 0-15
// carry K=0,1; lanes 16-31 carry K=2,3 -> one aligned float2 gather per lane
// per step. B = btw broadcast across the 16 columns (held in registers).
// D column 0 (lanes 0 and 16) holds the 16 scores; extracted with shuffles.
// EXEC is all-ones around every WMMA (tile loop is wave-uniform).
// ---------------------------------------------------------------------------
__global__ void eg_score_wmma(const float* __restrict__ embed,
                              const int*   __restrict__ eidx,
                              const float* __restrict__ btw,
                              float*       __restrict__ out,
                              int nTiles, int nEdges) {
    const int lane = threadIdx.x & 31;
    const int wave = threadIdx.x >> 5;
    const int half = lane >> 4;     // 0: K+0/K+1, 1: K+2/K+3
    const int m    = lane & 15;     // row M within the tile

    // B fragments: btw broadcast to all 16 columns. Per K-step each lane needs
    // (btw[4k + 2*half], btw[4k + 2*half + 1]). Loaded once, lives in VGPRs.
    v2f bfrag[32];
#pragma unroll
    for (int k = 0; k < 32; ++k) {
        bfrag[k] = *(const v2f*)(btw + 4 * k + 2 * half);
    }

    const int gwave  = blockIdx.x * (blockDim.x >> 5) + wave;
    const int nwaves = gridDim.x * (blockDim.x >> 5);

    for (int t = gwave; t < nTiles; t += nwaves) {
        const int base = t * 16;
        const int idx  = eidx[2 * (base + m)];      // column 0: A rows
        const float* rowp = embed + (size_t)idx * D_DIM + 2 * half;

        v8f acc = {};
#pragma unroll
        for (int k = 0; k < 32; ++k) {
            v2f a = *(const v2f*)(rowp + 4 * k);    // 8B aligned gather
            acc = __builtin_amdgcn_wmma_f32_16x16x4_f32(
                /*neg_a=*/false, a, /*neg_b=*/false, bfrag[k],
                /*c_mod=*/(short)0, acc, /*reuse_a=*/false, /*reuse_b=*/false);
        }

        // Extract column 0: D[r][0] in acc[r] of lane 0 (M=r) / lane 16 (M=r+8).
        float score = 0.0f;
#pragma unroll
        for (int r = 0; r < 8; ++r) {
            const float s0 = __shfl(acc[r], 0, 32);
            const float s8 = __shfl(acc[r], 16, 32);
            if (lane == r)     score = s0;
            if (lane == r + 8) score = s8;
        }

        if (lane < 16) {
            const float s = score;
            out[base + lane] = s / (s + EXP_NEG_EPS);
        }
    }

    // Tail edges (nEdges % 16 != 0 — not hit for 500000, kept for safety).
    const int tail = nTiles * 16;
    for (int e = tail + blockIdx.x * blockDim.x + threadIdx.x; e < nEdges;
         e += gridDim.x * blockDim.x) {
        const int idx = eidx[2 * e];
        const float* rp = embed + (size_t)idx * D_DIM;
        float s = 0.0f;
        for (int d = 0; d < D_DIM; ++d) s = fmaf(rp[d], btw[d], s);
        out[e] = s / (s + EXP_NEG_EPS);
    }
}

// ---------------------------------------------------------------------------
// Host launcher
// ---------------------------------------------------------------------------
extern "C" void kernel_launch(void* const* d_in, const int* in_sizes, int n_in,
                              void* d_out, int out_size, void* d_ws, size_t ws_size,
                              hipStream_t stream) {
    const float* embed  = (const float*)d_in[0];
    const float* weight = (const float*)d_in[1];
    const int*   eidx   = (const int*)d_in[2];
    float*       out    = (float*)d_out;
    float*       btw    = (float*)d_ws;            // 128 floats of scratch

    const int nEdges = in_sizes[1];                // weight has one entry per edge
    const int nTiles = nEdges / 16;

    eg_zero_btw<<<1, 128, 0, stream>>>(btw);

    eg_btw_accum<<<512, 256, 0, stream>>>(embed, weight, eidx, btw, nEdges);

    eg_score_wmma<<<512, 256, 0, stream>>>(embed, eidx, btw, out, nTiles, nEdges);
}